// pySlideLayer_352187319094
// MI455X (gfx1250) — compile-verified
//
#include <hip/hip_runtime.h>

// Problem constants (match reference)
#define B_      256
#define IN_DIM  4096
#define OUT_DIM 65536
#define IA      512
#define OA      2048

// L2 range-blocking: 8 slabs x 8192 rows x 16KB/row = 128 MB per slab (< 192 MB L2)
#define NP          8
#define RANGE_ROWS  (OUT_DIM / NP)

typedef __attribute__((ext_vector_type(2))) float v2f;
typedef __attribute__((ext_vector_type(4))) float v4f;
typedef __attribute__((ext_vector_type(8))) float v8f;

// ---------------------------------------------------------------------------
// Kernel 0: zero the per-(sample, slab) bucket counters
// ---------------------------------------------------------------------------
__global__ void zero_counts_kernel(int* __restrict__ counts) {
    int i = blockIdx.x * blockDim.x + threadIdx.x;
    if (i < B_ * NP) counts[i] = 0;
}

// ---------------------------------------------------------------------------
// Kernel 1: bucket each active output slot by which W-row slab it lives in.
// buckets[(b*NP+p)*OA + pos] = output slot o. Order within a bucket is
// arbitrary (atomic), but each output is computed independently later, so
// results are deterministic.
// ---------------------------------------------------------------------------
__global__ void bucket_kernel(const int* __restrict__ a_out,
                              int* __restrict__ counts,
                              int* __restrict__ buckets) {
    int gid = blockIdx.x * blockDim.x + threadIdx.x;
    if (gid >= B_ * OA) return;
    int b   = gid / OA;
    int o   = gid % OA;
    int row = a_out[gid];
    int p   = row / RANGE_ROWS;
    int pos = atomicAdd(&counts[b * NP + p], 1);
    buckets[(b * NP + p) * OA + pos] = o;
}

// ---------------------------------------------------------------------------
// Kernel 2: densify inputs. xdense[b, j] = sum of in_values[b,i] where
// active_in_indices[b,i] == j. Deterministic: each thread exclusively owns a
// 16-slot range of the dense vector and scans all 512 entries in order
// (no float atomics -> bit-reproducible).
// ---------------------------------------------------------------------------
__global__ __launch_bounds__(256)
void scatter_x_kernel(const float* __restrict__ in_values,
                      const int* __restrict__ a_in,
                      float* __restrict__ xdense) {
    __shared__ int   sidx[IA];
    __shared__ float sval[IA];
    __shared__ float xs[IN_DIM];
    int b = blockIdx.x;
    for (int i = threadIdx.x; i < IA; i += 256) {
        sidx[i] = a_in[(size_t)b * IA + i];
        sval[i] = in_values[(size_t)b * IA + i];
    }
    for (int i = threadIdx.x; i < IN_DIM; i += 256) xs[i] = 0.0f;
    __syncthreads();
    int owner = (int)threadIdx.x;             // owns dense slots [16*owner, 16*owner+16)
    for (int i = 0; i < IA; ++i) {
        int idx = sidx[i];
        if ((idx >> 4) == owner) xs[idx] += sval[i];
    }
    __syncthreads();
    const v4f* s = (const v4f*)xs;
    v4f*       d = (v4f*)(xdense + (size_t)b * IN_DIM);
    for (int i = threadIdx.x; i < IN_DIM / 4; i += 256) d[i] = s[i];
}

// ---------------------------------------------------------------------------
// Kernel 3 (launched NP times, sequentially): WMMA GEMV over one L2-resident
// W-row slab. Block = (slab tile-group x, sample y), 256 threads = 8 waves;
// each wave computes one 16-output tile with V_WMMA_F32_16X16X4_F32,
// K marching 4-at-a-time over the dense 4096 columns.
//
// A layout (ISA 16x4 f32): lane<16 -> M=lane, K in {0,1} across VGPR0/1;
// lane>=16 -> M=lane-16, K in {2,3}. We load float2 {W[r,4t+2kh], W[r,4t+2kh+1]}.
// B mirrors A; x is broadcast to all 16 N columns, so every accumulator
// column holds y[m]; lane0 holds y[0..7] in acc[0..7], lane16 holds y[8..15].
// ---------------------------------------------------------------------------
__global__ __launch_bounds__(256)
void gemv_pass_kernel(const float* __restrict__ W,
                      const float* __restrict__ bias,
                      const int*   __restrict__ a_out,
                      const float* __restrict__ xdense,
                      const int*   __restrict__ counts,
                      const int*   __restrict__ buckets,
                      float*       __restrict__ out,
                      int pass) {
    int b     = blockIdx.y;
    int count = counts[b * NP + pass];
    if ((int)blockIdx.x * 128 >= count) return;   // block-uniform early out

    __shared__ float xs[IN_DIM];
    {
        const v4f* src = (const v4f*)(xdense + (size_t)b * IN_DIM);
        v4f*       dst = (v4f*)xs;
        for (int i = threadIdx.x; i < IN_DIM / 4; i += 256) dst[i] = src[i];
    }
    __syncthreads();

    int wave = threadIdx.x >> 5;
    int lane = threadIdx.x & 31;
    int tileStart = (blockIdx.x * 8 + wave) * 16;
    if (tileStart >= count) return;               // wave-uniform

    int m  = lane & 15;
    int kh = lane >> 4;                           // which K-half this lane feeds

    const int* bkt = buckets + ((size_t)b * NP + pass) * OA;
    int slotIdx = tileStart + m;
    int valid   = slotIdx < count;
    int slot    = valid ? bkt[slotIdx] : bkt[tileStart];       // tail lanes: harmless dup
    int row     = a_out[(size_t)b * OA + slot];                // garbage rows never stored

    const float* wrow = W + (size_t)row * IN_DIM + 2 * kh;
    const float* xk   = xs + 2 * kh;

    v8f acc = {};
#pragma unroll 8
    for (int t = 0; t < IN_DIM / 4; ++t) {
        v2f a  = *(const v2f*)(wrow + 4 * t);     // A[m, 4t+2kh .. +1]
        v2f bb = *(const v2f*)(xk + 4 * t);       // x broadcast into B columns
        acc = __builtin_amdgcn_wmma_f32_16x16x4_f32(
            /*neg_a=*/false, a, /*neg_b=*/false, bb,
            /*c_mod=*/(short)0, acc, /*reuse_a=*/false, /*reuse_b=*/false);
    }

    // lane 0 -> tile rows 0..7, lane 16 -> tile rows 8..15 (columns all equal)
    if (lane == 0 || lane == 16) {
        int base = tileStart + (kh ? 8 : 0);
        for (int v = 0; v < 8; ++v) {
            int si = base + v;
            if (si < count) {
                int s = bkt[si];
                int r = a_out[(size_t)b * OA + s];
                out[(size_t)b * OA + s] = acc[v] + bias[r];
            }
        }
    }
}

// ---------------------------------------------------------------------------
// Launch
// ---------------------------------------------------------------------------
extern "C" void kernel_launch(void* const* d_in, const int* in_sizes, int n_in,
                              void* d_out, int out_size, void* d_ws, size_t ws_size,
                              hipStream_t stream) {
    const float* in_values = (const float*)d_in[0];
    const float* W         = (const float*)d_in[1];
    const float* bias      = (const float*)d_in[2];
    const int*   a_in      = (const int*)d_in[3];
    const int*   a_out     = (const int*)d_in[4];
    float*       out       = (float*)d_out;

    // Workspace layout: xdense (4 MB) | counts (8 KB) | buckets (16 MB)
    char*  ws      = (char*)d_ws;
    float* xdense  = (float*)ws;
    int*   counts  = (int*)(ws + (size_t)B_ * IN_DIM * sizeof(float));
    int*   buckets = (int*)(ws + (size_t)B_ * IN_DIM * sizeof(float)
                               + (size_t)B_ * NP * sizeof(int));

    zero_counts_kernel<<<(B_ * NP + 255) / 256, 256, 0, stream>>>(counts);
    bucket_kernel<<<(B_ * OA + 255) / 256, 256, 0, stream>>>(a_out, counts, buckets);
    scatter_x_kernel<<<B_, 256, 0, stream>>>(in_values, a_in, xdense);

    dim3 grid(OA / 128, B_);   // 16 tile-groups x 256 samples
    for (int p = 0; p < NP; ++p) {
        gemv_pass_kernel<<<grid, 256, 0, stream>>>(W, bias, a_out, xdense,
                                                   counts, buckets, out, p);
    }
}